// GCN_concat_readout_38654705664007
// MI455X (gfx1250) — compile-verified
//
#include <hip/hip_runtime.h>
#include <hip/hip_bf16.h>

// ---------------------------------------------------------------------------
// GCN (2x GCNConv + max/mean pool + MLP head) for MI455X / gfx1250.
//   - Node GEMMs (100000x128 @ 128x128) via V_WMMA_F32_16X16X4_F32 (exact f32)
//   - Edge scatter via global f32 atomics (L2-resident: 51MB << 192MB L2)
//   - Sorted-batch pooling with run-length flush (few atomics)
// ---------------------------------------------------------------------------

#define NNODES   100000
#define NEDGES   1600000
#define DIN      128
#define HIDW     128
#define NGRAPHS  64
#define NCLASSES 10

typedef __attribute__((ext_vector_type(2))) float v2f;
typedef __attribute__((ext_vector_type(8))) float v8f;

// -------------------------------- utility fills ----------------------------
__global__ void fill_f32_kernel(float* __restrict__ p, float v, long long n) {
  long long i = (long long)blockIdx.x * blockDim.x + threadIdx.x;
  long long stride = (long long)gridDim.x * blockDim.x;
  for (; i < n; i += stride) p[i] = v;
}

__global__ void fill_u32_kernel(unsigned* __restrict__ p, unsigned v, long long n) {
  long long i = (long long)blockIdx.x * blockDim.x + threadIdx.x;
  long long stride = (long long)gridDim.x * blockDim.x;
  for (; i < n; i += stride) p[i] = v;
}

// -------------------------------- degree / norm ----------------------------
__global__ void degree_kernel(const int* __restrict__ col, unsigned* __restrict__ cnt,
                              int n_edges) {
  int e = blockIdx.x * blockDim.x + threadIdx.x;
  if (e < n_edges) atomicAdd(&cnt[col[e]], 1u);
}

__global__ void dinv_kernel(const unsigned* __restrict__ cnt, float* __restrict__ dinv,
                            int n) {
  int i = blockIdx.x * blockDim.x + threadIdx.x;
  if (i < n) dinv[i] = rsqrtf((float)cnt[i]);  // cnt >= 1 (self-loop)
}

// -------------------------------- WMMA GEMM --------------------------------
// Y[n_rows,128] = X[n_rows,128] @ W[128,128].  n_rows multiple of 16.
// Block: 256 threads = 8 waves; wave w owns N-tile [16w,16w+16); 8 M-tiles/block.
// A (16x4 f32) layout: lane half h = lane>>4, m = lane&15; VGPR0=K(2h), VGPR1=K(2h+1)
//   -> contiguous float2 load from the node row.
// B (4x16 f32) layout mirrors A from LDS-resident W.
// C/D (16x16 f32): VGPR r -> M = r + 8h, N = lane&15.
__global__ __launch_bounds__(256)
void gemm_nodes_wmma(const float* __restrict__ X, const float* __restrict__ W,
                     float* __restrict__ Y, int n_mtiles) {
  __shared__ float sW[HIDW * HIDW];  // 64 KB (WGP has 320 KB LDS)
  for (int i = threadIdx.x; i < HIDW * HIDW; i += 256) sW[i] = W[i];
  __syncthreads();

  const int wave = threadIdx.x >> 5;
  const int lane = threadIdx.x & 31;
  const int half = lane >> 4;   // 0 or 1
  const int lm   = lane & 15;   // M (for A) / N (for B,C,D) index
  const int nc   = wave * 16 + lm;

  const int mtile0 = blockIdx.x * 8;
  for (int t = 0; t < 8; ++t) {
    const int mtile = mtile0 + t;
    if (mtile >= n_mtiles) break;
    const float* __restrict__ xrow = X + (long long)(mtile * 16 + lm) * DIN;

    v8f c = {0.f, 0.f, 0.f, 0.f, 0.f, 0.f, 0.f, 0.f};
#pragma unroll 8
    for (int s = 0; s < 32; ++s) {
      const int k0 = 4 * s + 2 * half;
      v2f a = *(const v2f*)(xrow + k0);            // {K=k0, K=k0+1} for this half
      v2f b;
      b.x = sW[k0 * HIDW + nc];
      b.y = sW[(k0 + 1) * HIDW + nc];
      c = __builtin_amdgcn_wmma_f32_16x16x4_f32(
          /*neg_a=*/false, a, /*neg_b=*/false, b,
          /*c_mod=*/(short)0, c, /*reuse_a=*/false, /*reuse_b=*/false);
    }
    float* __restrict__ yp = Y + (long long)(mtile * 16 + 8 * half) * HIDW + nc;
#pragma unroll
    for (int r = 0; r < 8; ++r) yp[(long long)r * HIDW] = c[r];
  }
}

// -------------------------------- edge scatter -----------------------------
// One wave per edge: gather 128 f32 of H[row], scale by dinv[row]*dinv[col],
// atomic-add into AGG[col].  4 floats (float4) per lane.
__global__ __launch_bounds__(256)
void edge_scatter_kernel(const int* __restrict__ row, const int* __restrict__ col,
                         const float* __restrict__ dinv, const float* __restrict__ H,
                         float* __restrict__ AGG, int n_edges) {
  const int wave = blockIdx.x * (blockDim.x >> 5) + (threadIdx.x >> 5);
  const int lane = threadIdx.x & 31;
  if (wave >= n_edges) return;
  const int r = row[wave];
  const int cdst = col[wave];
  const float nrm = dinv[r] * dinv[cdst];
  const float4 v = ((const float4*)(H + (long long)r * HIDW))[lane];
  float* dst = AGG + (long long)cdst * HIDW + lane * 4;
  atomicAdd(dst + 0, v.x * nrm);
  atomicAdd(dst + 1, v.y * nrm);
  atomicAdd(dst + 2, v.z * nrm);
  atomicAdd(dst + 3, v.w * nrm);
}

// Self-loop term + bias + ReLU, in place on AGG.
__global__ void node_finalize_kernel(float* __restrict__ AGG, const float* __restrict__ H,
                                     const float* __restrict__ dinv,
                                     const float* __restrict__ bias, long long n_elems) {
  long long idx = (long long)blockIdx.x * blockDim.x + threadIdx.x;
  if (idx >= n_elems) return;
  const int node = (int)(idx >> 7);
  const int c    = (int)(idx & 127);
  const float di = dinv[node];
  const float v  = AGG[idx] + di * di * H[idx] + bias[c];
  AGG[idx] = fmaxf(v, 0.f);
}

// -------------------------------- pooling ----------------------------------
// batch is sorted: each 128-thread block (thread = feature column) walks a
// strip of nodes keeping running sum/max per graph run; flushes on boundary.
#define POOL_STRIP 512
__global__ __launch_bounds__(128)
void pool_kernel(const float* __restrict__ H, const int* __restrict__ batch,
                 float* __restrict__ gsum, float* __restrict__ gmax,
                 float* __restrict__ gcnt, int n_nodes) {
  const int n  = threadIdx.x;  // feature column
  const int i0 = blockIdx.x * POOL_STRIP;
  const int i1 = min(i0 + POOL_STRIP, n_nodes);
  int cur = -1, run = 0;
  float s = 0.f, m = 0.f;
  for (int i = i0; i < i1; ++i) {
    const int g = batch[i];
    if (g != cur) {
      if (cur >= 0) {
        atomicAdd(&gsum[cur * HIDW + n], s);
        atomicMax((int*)&gmax[cur * HIDW + n], __float_as_int(m));  // values >= 0
        if (n == 0) atomicAdd(&gcnt[cur], (float)run);
      }
      cur = g; s = 0.f; m = 0.f; run = 0;
    }
    const float v = H[(long long)i * HIDW + n];
    s += v;
    m = fmaxf(m, v);
    ++run;
  }
  if (cur >= 0) {
    atomicAdd(&gsum[cur * HIDW + n], s);
    atomicMax((int*)&gmax[cur * HIDW + n], __float_as_int(m));
    if (n == 0) atomicAdd(&gcnt[cur], (float)run);
  }
}

// g = concat([gmax, gsum/cnt], axis=1)  -> [NGRAPHS, 256]
__global__ void pool_finalize_kernel(const float* __restrict__ gsum,
                                     const float* __restrict__ gmax,
                                     const float* __restrict__ gcnt,
                                     float* __restrict__ G) {
  int idx = blockIdx.x * blockDim.x + threadIdx.x;
  if (idx >= NGRAPHS * 2 * HIDW) return;
  const int g = idx >> 8;           // /256
  const int c = idx & 255;
  G[idx] = (c < HIDW) ? gmax[g * HIDW + c]
                      : gsum[g * HIDW + (c - HIDW)] / fmaxf(gcnt[g], 1.f);
}

// -------------------------------- MLP head ---------------------------------
__global__ void dense_relu_kernel(const float* __restrict__ A, const float* __restrict__ W,
                                  const float* __restrict__ b, float* __restrict__ O,
                                  int G, int K, int N) {
  int idx = blockIdx.x * blockDim.x + threadIdx.x;
  if (idx >= G * N) return;
  const int g = idx / N, n = idx % N;
  float acc = b[n];
  for (int k = 0; k < K; ++k) acc = fmaf(A[g * K + k], W[k * N + n], acc);
  O[idx] = fmaxf(acc, 0.f);
}

__global__ void head_logsoftmax_kernel(const float* __restrict__ t2,
                                       const float* __restrict__ W,
                                       const float* __restrict__ b,
                                       float* __restrict__ out) {
  const int g = threadIdx.x;
  if (g >= NGRAPHS) return;
  float logits[NCLASSES];
#pragma unroll
  for (int j = 0; j < NCLASSES; ++j) logits[j] = b[j];
  for (int k = 0; k < 64; ++k) {
    const float v = t2[g * 64 + k];
#pragma unroll
    for (int j = 0; j < NCLASSES; ++j) logits[j] = fmaf(v, W[k * NCLASSES + j], logits[j]);
  }
  float m = logits[0];
#pragma unroll
  for (int j = 1; j < NCLASSES; ++j) m = fmaxf(m, logits[j]);
  float s = 0.f;
#pragma unroll
  for (int j = 0; j < NCLASSES; ++j) s += expf(logits[j] - m);
  const float lse = m + logf(s);
#pragma unroll
  for (int j = 0; j < NCLASSES; ++j) out[g * NCLASSES + j] = logits[j] - lse;
}

// -------------------------------- launcher ---------------------------------
extern "C" void kernel_launch(void* const* d_in, const int* in_sizes, int n_in,
                              void* d_out, int out_size, void* d_ws, size_t ws_size,
                              hipStream_t stream) {
  const float* x    = (const float*)d_in[0];
  const int*   ei   = (const int*)d_in[1];   // [2, NEDGES] row-major
  const int*   batch= (const int*)d_in[2];
  const float* W1   = (const float*)d_in[3];
  const float* b1   = (const float*)d_in[4];
  const float* W2   = (const float*)d_in[5];
  const float* b2   = (const float*)d_in[6];
  const float* Wl1  = (const float*)d_in[7];
  const float* bl1  = (const float*)d_in[8];
  const float* Wl2  = (const float*)d_in[9];
  const float* bl2  = (const float*)d_in[10];
  const float* Wl3  = (const float*)d_in[11];
  const float* bl3  = (const float*)d_in[12];
  float* out = (float*)d_out;

  const int* erow = ei;            // source j
  const int* ecol = ei + NEDGES;   // target i

  // ---- workspace layout (256B aligned) ----
  char* ws = (char*)d_ws;
  size_t off = 0;
  auto take = [&](size_t bytes) -> char* {
    char* p = ws + off;
    off = (off + bytes + 255) & ~(size_t)255;
    return p;
  };
  const size_t SZ_H = (size_t)NNODES * HIDW * sizeof(float);  // 51.2 MB
  float*    hA   = (float*)take(SZ_H);
  float*    hB   = (float*)take(SZ_H);
  float*    dinv = (float*)take((size_t)NNODES * sizeof(float));
  unsigned* cnt  = (unsigned*)take((size_t)NNODES * sizeof(unsigned));
  float*    gsum = (float*)take((size_t)NGRAPHS * HIDW * sizeof(float));
  float*    gmax = (float*)take((size_t)NGRAPHS * HIDW * sizeof(float));
  float*    gcnt = (float*)take((size_t)NGRAPHS * sizeof(float));
  float*    Gmat = (float*)take((size_t)NGRAPHS * 2 * HIDW * sizeof(float));
  float*    t1   = (float*)take((size_t)NGRAPHS * HIDW * sizeof(float));
  float*    t2   = (float*)take((size_t)NGRAPHS * 64 * sizeof(float));

  const long long NE = (long long)NNODES * HIDW;  // 12.8M feature elements
  const int n_mtiles = NNODES / 16;               // 6250 (exact)

  // ---- degrees & symmetric norm ----
  fill_u32_kernel<<<1024, 256, 0, stream>>>(cnt, 1u, NNODES);  // self-loops
  degree_kernel<<<(NEDGES + 255) / 256, 256, 0, stream>>>(ecol, cnt, NEDGES);
  dinv_kernel<<<(NNODES + 255) / 256, 256, 0, stream>>>(cnt, dinv, NNODES);

  // ---- layer 1: hA = x @ W1 ; agg -> hB ; relu(hB + dinv^2*hA + b1) ----
  gemm_nodes_wmma<<<(n_mtiles + 7) / 8, 256, 0, stream>>>(x, W1, hA, n_mtiles);
  fill_f32_kernel<<<2048, 256, 0, stream>>>(hB, 0.f, NE);
  edge_scatter_kernel<<<NEDGES / 8, 256, 0, stream>>>(erow, ecol, dinv, hA, hB, NEDGES);
  node_finalize_kernel<<<(int)((NE + 255) / 256), 256, 0, stream>>>(hB, hA, dinv, b1, NE);

  // ---- layer 2: hA = hB @ W2 ; agg -> hB ; relu(hB + dinv^2*hA + b2) ----
  gemm_nodes_wmma<<<(n_mtiles + 7) / 8, 256, 0, stream>>>(hB, W2, hA, n_mtiles);
  fill_f32_kernel<<<2048, 256, 0, stream>>>(hB, 0.f, NE);
  edge_scatter_kernel<<<NEDGES / 8, 256, 0, stream>>>(erow, ecol, dinv, hA, hB, NEDGES);
  node_finalize_kernel<<<(int)((NE + 255) / 256), 256, 0, stream>>>(hB, hA, dinv, b2, NE);

  // ---- pooling ----
  fill_f32_kernel<<<32, 256, 0, stream>>>(gsum, 0.f, NGRAPHS * HIDW);
  fill_f32_kernel<<<32, 256, 0, stream>>>(gmax, 0.f, NGRAPHS * HIDW);
  fill_f32_kernel<<<1, 64, 0, stream>>>(gcnt, 0.f, NGRAPHS);
  pool_kernel<<<(NNODES + POOL_STRIP - 1) / POOL_STRIP, 128, 0, stream>>>(
      hB, batch, gsum, gmax, gcnt, NNODES);
  pool_finalize_kernel<<<(NGRAPHS * 2 * HIDW + 255) / 256, 256, 0, stream>>>(
      gsum, gmax, gcnt, Gmat);

  // ---- MLP head + log_softmax ----
  dense_relu_kernel<<<(NGRAPHS * HIDW + 255) / 256, 256, 0, stream>>>(
      Gmat, Wl1, bl1, t1, NGRAPHS, 2 * HIDW, HIDW);
  dense_relu_kernel<<<(NGRAPHS * 64 + 255) / 256, 256, 0, stream>>>(
      t1, Wl2, bl2, t2, NGRAPHS, HIDW, 64);
  head_logsoftmax_kernel<<<1, 64, 0, stream>>>(t2, Wl3, bl3, out);

  (void)in_sizes; (void)n_in; (void)out_size; (void)ws_size;
}